// MAF_Extractor_36223754174722
// MI455X (gfx1250) — compile-verified
//
#include <hip/hip_runtime.h>

typedef __attribute__((ext_vector_type(2))) float v2f;
typedef __attribute__((ext_vector_type(8))) float v8f;

#define NPTS 6890
#define CH   256
#define HH   56
#define WW   56
#define HW   (HH*WW)
#define NT   32          // points per block tile
#define C0   128         // layer0 out channels
#define C1   64          // layer1 out channels
#define C2   5           // layer2 out channels
#define NEG_SLOPE 0.01f

__global__ __launch_bounds__(256)
void maf_fused_kernel(const float* __restrict__ pts,   // [B,N,2]
                      const float* __restrict__ im,    // [B,256,56,56]
                      const float* __restrict__ w0, const float* __restrict__ b0,  // [128,256],[128]
                      const float* __restrict__ w1, const float* __restrict__ b1,  // [64,384],[64]
                      const float* __restrict__ w2, const float* __restrict__ b2,  // [5,320],[5]
                      float* __restrict__ out_mesh,    // [B,5,N] flat
                      float* __restrict__ out_pf)      // [B,256,N] flat
{
    __shared__ float pf_lds[CH * NT];   // 32 KB, [c][p]
    __shared__ float y0_lds[C0 * NT];   // 16 KB
    __shared__ float y1_lds[C1 * NT];   //  8 KB

    const int t  = threadIdx.x;
    const int b  = blockIdx.y;
    const int n0 = blockIdx.x * NT;

    // ---------------- Phase 1: bilinear gather (grid_sample, align_corners=True) --------
    {
        const int p  = t & (NT - 1);          // point within tile
        const int cg = t >> 5;                // wave id = channel group (32 ch each)
        const int n  = n0 + p;
        const bool valid = (n < NPTS);

        float px = 0.f, py = 0.f;
        if (valid) {
            px = pts[((size_t)b * NPTS + n) * 2 + 0];
            py = pts[((size_t)b * NPTS + n) * 2 + 1];
        }
        float x = (px + 1.f) * 0.5f * (float)(WW - 1);
        float y = (py + 1.f) * 0.5f * (float)(HH - 1);
        float fx0 = floorf(x), fy0 = floorf(y);
        float wx1 = x - fx0, wy1 = y - fy0;
        float wx0 = 1.f - wx1, wy0 = 1.f - wy1;
        int x0 = (int)fx0, y0 = (int)fy0;
        int x1 = x0 + 1,   y1 = y0 + 1;
        float vx0 = (x0 >= 0 && x0 < WW) ? 1.f : 0.f;
        float vx1 = (x1 >= 0 && x1 < WW) ? 1.f : 0.f;
        float vy0 = (y0 >= 0 && y0 < HH) ? 1.f : 0.f;
        float vy1 = (y1 >= 0 && y1 < HH) ? 1.f : 0.f;
        int cx0 = min(max(x0, 0), WW - 1), cx1 = min(max(x1, 0), WW - 1);
        int cy0 = min(max(y0, 0), HH - 1), cy1 = min(max(y1, 0), HH - 1);
        float w00 = wy0 * wx0 * vy0 * vx0;
        float w01 = wy0 * wx1 * vy0 * vx1;
        float w10 = wy1 * wx0 * vy1 * vx0;
        float w11 = wy1 * wx1 * vy1 * vx1;
        if (!valid) { w00 = w01 = w10 = w11 = 0.f; }
        const int o00 = cy0 * WW + cx0, o01 = cy0 * WW + cx1;
        const int o10 = cy1 * WW + cx0, o11 = cy1 * WW + cx1;

        const float* imb = im + ((size_t)b * CH + cg * 32) * HW;
        float* pfo = out_pf + ((size_t)b * CH + cg * 32) * NPTS + n;

        #pragma unroll 4
        for (int c = 0; c < 32; ++c) {
            const float* base = imb + (size_t)c * HW;
            float v = w00 * base[o00] + w01 * base[o01]
                    + w10 * base[o10] + w11 * base[o11];
            pf_lds[(cg * 32 + c) * NT + p] = v;
            if (valid) pfo[(size_t)c * NPTS] = v;   // coalesced per-wave across lanes
        }
    }
    __syncthreads();

    // WMMA fp32 fragment addressing (ISA 7.12.2):
    //  A 16x4:  lane l -> row M = l&15, VGPR{0,1} = K = 2*(l>>4) + {0,1}
    //  B 4x16:  lane l -> col N = l&15, VGPR{0,1} = K = 2*(l>>4) + {0,1}
    //  C/D:     lane l, VGPR v -> M = v + 8*(l>>4), N = l&15
    const int lane = t & 31;
    const int wv   = t >> 5;
    const int hi   = lane >> 4;
    const int lm   = lane & 15;
    const int kb   = hi * 2;

    // ---------------- Phase 2: layer0  y0 = leaky(W0[128,256] @ pf[256,32] + b0) --------
    for (int pair = wv; pair < 16; pair += 8) {     // 8 mtiles x 2 ntiles
        const int mt = pair >> 1;
        const int nt = pair & 1;
        const int col = nt * 16 + lm;
        v8f acc;
        #pragma unroll
        for (int v = 0; v < 8; ++v) acc[v] = b0[mt * 16 + v + 8 * hi];
        const float* wrow = w0 + (size_t)(mt * 16 + lm) * CH;
        for (int k = 0; k < CH; k += 4) {
            v2f a = *(const v2f*)(wrow + k + kb);
            v2f bm;
            bm.x = pf_lds[(k + kb    ) * NT + col];
            bm.y = pf_lds[(k + kb + 1) * NT + col];
            acc = __builtin_amdgcn_wmma_f32_16x16x4_f32(
                false, a, false, bm, (short)0, acc, false, false);
        }
        #pragma unroll
        for (int v = 0; v < 8; ++v) {
            float r = acc[v];
            r = (r > 0.f) ? r : NEG_SLOPE * r;
            y0_lds[(mt * 16 + v + 8 * hi) * NT + col] = r;
        }
    }
    __syncthreads();

    // ---------------- Phase 3: layer1  y1 = leaky(W1[64,384] @ [y0;pf] + b1) ------------
    {
        const int mt = wv >> 1;                     // 4 mtiles x 2 ntiles = 8 pairs
        const int nt = wv & 1;
        const int col = nt * 16 + lm;
        v8f acc;
        #pragma unroll
        for (int v = 0; v < 8; ++v) acc[v] = b1[mt * 16 + v + 8 * hi];
        const float* wrow = w1 + (size_t)(mt * 16 + lm) * (C0 + CH);
        for (int k = 0; k < C0; k += 4) {           // first 128 in-ch: y0
            v2f a = *(const v2f*)(wrow + k + kb);
            v2f bm;
            bm.x = y0_lds[(k + kb    ) * NT + col];
            bm.y = y0_lds[(k + kb + 1) * NT + col];
            acc = __builtin_amdgcn_wmma_f32_16x16x4_f32(
                false, a, false, bm, (short)0, acc, false, false);
        }
        for (int k = 0; k < CH; k += 4) {           // next 256 in-ch: pf (skip concat)
            v2f a = *(const v2f*)(wrow + C0 + k + kb);
            v2f bm;
            bm.x = pf_lds[(k + kb    ) * NT + col];
            bm.y = pf_lds[(k + kb + 1) * NT + col];
            acc = __builtin_amdgcn_wmma_f32_16x16x4_f32(
                false, a, false, bm, (short)0, acc, false, false);
        }
        #pragma unroll
        for (int v = 0; v < 8; ++v) {
            float r = acc[v];
            r = (r > 0.f) ? r : NEG_SLOPE * r;
            y1_lds[(mt * 16 + v + 8 * hi) * NT + col] = r;
        }
    }
    __syncthreads();

    // ---------------- Phase 4: layer2 (Cout=5) scalar + ReLU ----------------------------
    if (t < C2 * NT) {                              // 160 threads
        const int o = t >> 5;                       // 0..4
        const int p = t & 31;
        float acc = b2[o];
        const float* wr = w2 + (size_t)o * (C1 + CH);
        #pragma unroll 8
        for (int c = 0; c < C1; ++c) acc += wr[c] * y1_lds[c * NT + p];
        #pragma unroll 8
        for (int c = 0; c < CH; ++c) acc += wr[C1 + c] * pf_lds[c * NT + p];
        acc = fmaxf(acc, 0.f);                      // last_op ReLU
        const int n = n0 + p;
        if (n < NPTS)
            out_mesh[(size_t)b * (C2 * NPTS) + (size_t)o * NPTS + n] = acc;
    }
}

extern "C" void kernel_launch(void* const* d_in, const int* in_sizes, int n_in,
                              void* d_out, int out_size, void* d_ws, size_t ws_size,
                              hipStream_t stream) {
    const float* pts = (const float*)d_in[0];
    const float* im  = (const float*)d_in[1];
    const float* w0  = (const float*)d_in[2];
    const float* b0  = (const float*)d_in[3];
    const float* w1  = (const float*)d_in[4];
    const float* b1  = (const float*)d_in[5];
    const float* w2  = (const float*)d_in[6];
    const float* b2  = (const float*)d_in[7];

    float* out      = (float*)d_out;
    float* out_mesh = out;                              // [64, 5*6890] flat
    float* out_pf   = out + (size_t)64 * C2 * NPTS;     // [64, 256, 6890] flat

    dim3 grid((NPTS + NT - 1) / NT, 64);
    maf_fused_kernel<<<grid, 256, 0, stream>>>(pts, im, w0, b0, w1, b1, w2, b2,
                                               out_mesh, out_pf);
}